// Attention_73607149519658
// MI455X (gfx1250) — compile-verified
//
#include <hip/hip_runtime.h>
#include <hip/hip_bf16.h>
#include <math.h>

// Problem dims (fixed by the reference)
#define BB 32
#define TT 2048
#define HH 512
#define TSPLIT 8            // context-vector partial split along T
#define TCHUNK (TT / TSPLIT)

typedef __attribute__((ext_vector_type(2))) float v2f;
typedef __attribute__((ext_vector_type(8))) float v8f;

// ---------------------------------------------------------------------------
// Kernel 1: dec_fea[B,H] = s_t_hat[B,H] @ W_dec[H,H] + b_dec  via
// V_WMMA_F32_16X16X4_F32 (exact f32 matrix path on CDNA5).
// One wave per 16x16 output tile: 2 M-tiles x 32 N-tiles = 64 waves.
// VGPR layouts per CDNA5 ISA 7.12.2:
//   A (16x4): lanes 0-15 -> K={0,1}, lanes 16-31 -> K={2,3}, M = lane%16
//   B (4x16): lanes 0-15 -> K={0,1}, lanes 16-31 -> K={2,3}, N = lane%16
//   C/D     : VGPR r, lanes 0-15 -> M=r, lanes 16-31 -> M=r+8, N = lane%16
// ---------------------------------------------------------------------------
__global__ __launch_bounds__(32) void dec_proj_wmma(
    const float* __restrict__ S,      // [B,H]
    const float* __restrict__ W,      // [H,H]
    const float* __restrict__ bias,   // [H]
    float* __restrict__ dec)          // [B,H]
{
  const int wave = blockIdx.x;            // 0..63
  const int m0   = (wave >> 5) << 4;      // 0 or 16
  const int n0   = (wave & 31) << 4;      // 0..496
  const int l    = threadIdx.x;
  const int half = l >> 4;                // 0 or 1
  const int lm   = l & 15;

  v8f acc = {};
  for (int k = 0; k < HH; k += 4) {
    const int ka = k + 2 * half;          // K index pair owned by this lane half
    v2f a, b;
    a.x = S[(m0 + lm) * HH + ka];
    a.y = S[(m0 + lm) * HH + ka + 1];
    b.x = W[(ka)     * HH + n0 + lm];
    b.y = W[(ka + 1) * HH + n0 + lm];
    acc = __builtin_amdgcn_wmma_f32_16x16x4_f32(
        /*neg_a=*/false, a, /*neg_b=*/false, b,
        /*c_mod=*/(short)0, acc, /*reuse_a=*/false, /*reuse_b=*/false);
  }
  #pragma unroll
  for (int r = 0; r < 8; ++r) {
    const int row = m0 + r + 8 * half;
    const int col = n0 + lm;
    dec[row * HH + col] = acc[r] + bias[col];
  }
}

// ---------------------------------------------------------------------------
// Kernel 2: scores[b,t] = sum_h tanh(ef[b,t,h] + dec[b,h] + cov[b,t]*Wc[h])*v[h]
// One wave per (b,t) row; each lane streams 16 h's as float4 (fully coalesced
// 2KB per row). This pass reads the 134MB encoder_feature once -> HBM bound.
// ---------------------------------------------------------------------------
__global__ __launch_bounds__(256) void scores_kernel(
    const float* __restrict__ ef,     // [B*T, H]
    const float* __restrict__ dec,    // [B, H]
    const float* __restrict__ cov,    // [B, T]
    const float* __restrict__ Wc,     // [H]
    const float* __restrict__ vvec,   // [H]
    float* __restrict__ scores)       // [B*T]
{
  const int lane = threadIdx.x & 31;
  const int wv   = threadIdx.x >> 5;
  const int row  = blockIdx.x * 8 + wv;       // b*T + t  (< 65536)
  const int b    = row >> 11;                 // /TT
  const float* erow = ef + row * HH;
  const float* drow = dec + b * HH;
  const float  c    = cov[row];

  float s = 0.f;
  #pragma unroll
  for (int i = 0; i < 4; ++i) {
    const int h = i * 128 + lane * 4;
    const float4 e4 = *(const float4*)(erow + h);
    const float4 d4 = *(const float4*)(drow + h);
    const float4 w4 = *(const float4*)(Wc + h);
    const float4 v4 = *(const float4*)(vvec + h);
    s += tanhf(e4.x + d4.x + c * w4.x) * v4.x;
    s += tanhf(e4.y + d4.y + c * w4.y) * v4.y;
    s += tanhf(e4.z + d4.z + c * w4.z) * v4.z;
    s += tanhf(e4.w + d4.w + c * w4.w) * v4.w;
  }
  #pragma unroll
  for (int off = 16; off >= 1; off >>= 1)
    s += __shfl_xor(s, off, 32);
  if (lane == 0) scores[row] = s;
}

// ---------------------------------------------------------------------------
// Kernel 3: masked softmax + renorm + coverage update. One block per batch b.
// attn = exp(s - max) * mask / sum(exp(s - max) * mask)  (mask-renorm cancels
// the unmasked softmax denominator exactly, matching the reference).
// ---------------------------------------------------------------------------
__global__ __launch_bounds__(256) void softmax_kernel(
    const float* __restrict__ scores,  // [B,T]
    const float* __restrict__ mask,    // [B,T]
    const float* __restrict__ cov,     // [B,T]
    float* __restrict__ attn,          // [B,T]
    float* __restrict__ cov_new)       // [B,T]
{
  __shared__ float red[8];
  const int b = blockIdx.x;
  const int tid = threadIdx.x, lane = tid & 31, wv = tid >> 5;
  const float* srow = scores + b * TT;

  // --- row max ---
  float m = -3.402823466e38f;
  float sv[TT / 256];
  #pragma unroll
  for (int i = 0; i < TT / 256; ++i) {
    sv[i] = srow[tid + i * 256];
    m = fmaxf(m, sv[i]);
  }
  #pragma unroll
  for (int off = 16; off >= 1; off >>= 1)
    m = fmaxf(m, __shfl_xor(m, off, 32));
  if (lane == 0) red[wv] = m;
  __syncthreads();
  #pragma unroll
  for (int i = 0; i < 8; ++i) m = fmaxf(m, red[i]);
  __syncthreads();

  // --- masked exp + sum ---
  float p[TT / 256];
  float acc = 0.f;
  #pragma unroll
  for (int i = 0; i < TT / 256; ++i) {
    const int t = tid + i * 256;
    p[i] = expf(sv[i] - m) * mask[b * TT + t];
    acc += p[i];
  }
  #pragma unroll
  for (int off = 16; off >= 1; off >>= 1)
    acc += __shfl_xor(acc, off, 32);
  if (lane == 0) red[wv] = acc;
  __syncthreads();
  float tot = 0.f;
  #pragma unroll
  for (int i = 0; i < 8; ++i) tot += red[i];
  const float inv = 1.0f / tot;

  // --- write attn and coverage_new ---
  #pragma unroll
  for (int i = 0; i < TT / 256; ++i) {
    const int t = tid + i * 256;
    const float a = p[i] * inv;
    attn[b * TT + t]    = a;
    cov_new[b * TT + t] = cov[b * TT + t] + a;
  }
}

// ---------------------------------------------------------------------------
// Kernel 4: context-vector partials. Grid (B, TSPLIT), 512 threads = H lanes.
// Each block streams a 256-timestep chunk of encoder_outputs (2KB/row,
// coalesced), attn chunk cached in LDS. global_prefetch_b8 pulls rows 4
// timesteps ahead (speculative OOB prefetch past buffer end is dropped by HW).
// ---------------------------------------------------------------------------
__global__ __launch_bounds__(512) void ctx_partial(
    const float* __restrict__ enc,    // [B,T,H]
    const float* __restrict__ attn,   // [B,T]
    float* __restrict__ part)         // [B, TSPLIT, H]
{
  __shared__ float a_s[TCHUNK];
  const int b = blockIdx.x, s = blockIdx.y;
  const int h = threadIdx.x;
  const int t0 = s * TCHUNK;
  const float* e = enc + (b * TT + t0) * HH;

  if (threadIdx.x < TCHUNK)
    a_s[threadIdx.x] = attn[b * TT + t0 + threadIdx.x];
  __syncthreads();

  float acc = 0.f;
  for (int t = 0; t < TCHUNK; ++t) {
    __builtin_prefetch(e + (t + 4) * HH + h, 0, 0);   // global_prefetch_b8
    acc = fmaf(a_s[t], e[t * HH + h], acc);
  }
  part[(b * TSPLIT + s) * HH + h] = acc;
}

// Kernel 5: deterministic reduce of partials -> c_t[B,H]
__global__ __launch_bounds__(512) void ctx_reduce(
    const float* __restrict__ part, float* __restrict__ c_t)
{
  const int b = blockIdx.x, h = threadIdx.x;
  float s = 0.f;
  #pragma unroll
  for (int i = 0; i < TSPLIT; ++i)
    s += part[(b * TSPLIT + i) * HH + h];
  c_t[b * HH + h] = s;
}

// ---------------------------------------------------------------------------
extern "C" void kernel_launch(void* const* d_in, const int* in_sizes, int n_in,
                              void* d_out, int out_size, void* d_ws, size_t ws_size,
                              hipStream_t stream) {
  const float* s_t_hat  = (const float*)d_in[0];   // [B,H]
  const float* enc_out  = (const float*)d_in[1];   // [B,T,H]
  const float* enc_feat = (const float*)d_in[2];   // [B*T,H]
  const float* mask     = (const float*)d_in[3];   // [B,T]
  const float* cov      = (const float*)d_in[4];   // [B,T]
  const float* W_dec    = (const float*)d_in[5];   // [H,H]
  const float* b_dec    = (const float*)d_in[6];   // [H]
  const float* W_c      = (const float*)d_in[7];   // [1,H] -> H
  const float* v        = (const float*)d_in[8];   // [H,1] -> H

  float* out     = (float*)d_out;
  float* c_t     = out;                    // [B,H]
  float* attn    = out + BB * HH;          // [B,T]
  float* cov_new = attn + BB * TT;         // [B,T]

  float* ws     = (float*)d_ws;
  float* dec    = ws;                      // B*H      = 16384 floats
  float* scores = dec + BB * HH;           // B*T      = 65536 floats
  float* part   = scores + BB * TT;        // B*8*H    = 131072 floats (~832KB total)

  dec_proj_wmma<<<64, 32, 0, stream>>>(s_t_hat, W_dec, b_dec, dec);
  scores_kernel<<<BB * TT / 8, 256, 0, stream>>>(enc_feat, dec, cov, W_c, v, scores);
  softmax_kernel<<<BB, 256, 0, stream>>>(scores, mask, cov, attn, cov_new);
  ctx_partial<<<dim3(BB, TSPLIT), 512, 0, stream>>>(enc_out, attn, part);
  ctx_reduce<<<BB, 512, 0, stream>>>(part, c_t);
}